// IterNormRotation_11527692222571
// MI455X (gfx1250) — compile-verified
//
#include <hip/hip_runtime.h>
#include <cstdint>

// Problem constants (from reference): X [64,128,56,56] fp32, rot [1,128,128] fp32
#define B_    64
#define C_CH  128
#define HW_   3136            // 56*56
#define M_TOT (B_ * HW_)      // 200704
#define CHUNK 64              // m-columns per LDS tile
#define XS    68              // LDS row stride for x tiles (floats, 16B aligned, conflict-reducing)
#define MS    132             // LDS row stride for 128x128 matrices
#define NEWTON_T 10
#define EPS_  1e-5f

typedef __attribute__((ext_vector_type(2))) float v2f;
typedef __attribute__((ext_vector_type(8))) float v8f;
typedef __attribute__((ext_vector_type(4))) int   v4i;

#if __has_builtin(__builtin_amdgcn_wmma_f32_16x16x4_f32)
#define HAVE_WMMA_F32 1
#endif
#if __has_builtin(__builtin_amdgcn_global_load_async_to_lds_b128)
#define HAVE_ASYNC_LDS 1
#endif

__device__ __forceinline__ v8f v8f_zero() {
  v8f z;
#pragma unroll
  for (int i = 0; i < 8; ++i) z[i] = 0.0f;
  return z;
}

// D = A(16x4) * B(4x16) + C, fp32 WMMA
__device__ __forceinline__ v8f wmma_f32_k4(v2f a, v2f b, v8f c) {
#if defined(HAVE_WMMA_F32)
  return __builtin_amdgcn_wmma_f32_16x16x4_f32(false, a, false, b, (short)0, c,
                                               false, false);
#else
  c[0] += a.x * b.x;  // placeholder so compile succeeds; histogram will show wmma=0
  return c;
#endif
}

// 16B global -> LDS async copy (per lane)
__device__ __forceinline__ void async_cp16(float* lds_dst, const float* g_src) {
#if defined(HAVE_ASYNC_LDS)
  __builtin_amdgcn_global_load_async_to_lds_b128((v4i*)g_src, (v4i*)lds_dst, 0, 0);
#else
  *(float4*)lds_dst = *(const float4*)g_src;
#endif
}

__device__ __forceinline__ void wait_async_copies() {
#if defined(HAVE_ASYNC_LDS)
#if __has_builtin(__builtin_amdgcn_s_wait_asynccnt)
  __builtin_amdgcn_s_wait_asynccnt(0);
#else
  asm volatile("s_wait_asynccnt 0" ::: "memory");
#endif
#endif
}

// --- WMMA fragment loaders (ISA 7.12.2 layouts, wave32) ---
// A 16x4 fp32: lanes 0-15 hold rows, {v0,v1}={K0,K1}; lanes 16-31: {K2,K3}
__device__ __forceinline__ v2f loadA(const float* Mx, int stride, int row0,
                                     int k0, int lane) {
  int r  = row0 + (lane & 15);
  int kk = k0 + ((lane >> 4) << 1);
  v2f a;
  a.x = Mx[r * stride + kk];
  a.y = Mx[r * stride + kk + 1];
  return a;
}
// B 4x16 fp32 from row-major [k][n] storage: v0 = rows {k0,k0+1} over lane halves, v1 = {k0+2,k0+3}
__device__ __forceinline__ v2f loadB_rm(const float* Mx, int stride, int k0,
                                        int col0, int lane) {
  int n  = col0 + (lane & 15);
  int kk = k0 + (lane >> 4);
  v2f b;
  b.x = Mx[kk * stride + n];
  b.y = Mx[(kk + 2) * stride + n];
  return b;
}
// B 4x16 where B[k][j] = T[j][k] (transposed storage, used for Gram = x * x^T)
__device__ __forceinline__ v2f loadB_tr(const float* Mx, int stride, int k0,
                                        int col0, int lane) {
  int n  = col0 + (lane & 15);
  int kk = k0 + (lane >> 4);
  v2f b;
  b.x = Mx[n * stride + kk];
  b.y = Mx[n * stride + kk + 2];
  return b;
}
// C/D 16x16 fp32: VGPR r -> rows {r, r+8} over lane halves, col = lane&15
__device__ __forceinline__ void storeC(float* Mx, int stride, int row0,
                                       int col0, int lane, v8f c) {
#pragma unroll
  for (int r = 0; r < 8; ++r) {
    int row = row0 + r + ((lane >> 4) << 3);
    Mx[row * stride + col0 + (lane & 15)] = c[r];
  }
}

// ---------------- Kernel 1: partial Gram (x * x^T) + channel sums ----------------
// grid: 128 blocks (one per (batch, half)); block: 256 threads = 8 waves.
// Each wave owns a 16x128 row-slab of the 128x128 Gram accumulator.
__global__ void __launch_bounds__(256) k_gram(const float* __restrict__ X,
                                              float* __restrict__ gpart,
                                              float* __restrict__ spart) {
  __shared__ float xt[C_CH * XS];
  const int p    = blockIdx.x;
  const int b    = p >> 1;
  const int half = p & 1;
  const int c0   = half ? 25 : 0;
  const int nch  = half ? 24 : 25;  // 25+24 = 49 chunks of 64 = 3136
  const int tid  = threadIdx.x;
  const int lane = tid & 31;
  const int wv   = tid >> 5;

  v8f acc[8];
#pragma unroll
  for (int t = 0; t < 8; ++t) acc[t] = v8f_zero();
  float ssum = 0.0f;

  const float* Xb  = X + (size_t)b * C_CH * HW_;
  const int    row = tid >> 1;         // 2 threads per channel row
  const int    mo  = (tid & 1) * 32;   // each covers 32 contiguous floats

  for (int ch = 0; ch < nch; ++ch) {
    const int    hw0 = (c0 + ch) * CHUNK;
    const float* g   = Xb + (size_t)row * HW_ + hw0 + mo;
    float*       l   = &xt[row * XS + mo];
#pragma unroll
    for (int q = 0; q < 8; ++q) async_cp16(l + q * 4, g + q * 4);
    if (ch + 1 < nch)
      __builtin_prefetch(Xb + (size_t)row * HW_ + hw0 + CHUNK + mo, 0, 0);
    wait_async_copies();
    __syncthreads();

    if (tid < C_CH) {  // uniform per-wave: waves 0-3 fully active
      const float* r = &xt[tid * XS];
#pragma unroll 8
      for (int j = 0; j < CHUNK; ++j) ssum += r[j];
    }

    for (int k = 0; k < CHUNK; k += 4) {
      v2f a = loadA(xt, XS, wv * 16, k, lane);
#pragma unroll
      for (int tj = 0; tj < 8; ++tj) {
        v2f bf   = loadB_tr(xt, XS, k, tj * 16, lane);
        acc[tj]  = wmma_f32_k4(a, bf, acc[tj]);
      }
    }
    __syncthreads();
  }

  float* gp = gpart + (size_t)p * (C_CH * C_CH);
#pragma unroll
  for (int tj = 0; tj < 8; ++tj)
#pragma unroll
    for (int r = 0; r < 8; ++r) {
      int grow = wv * 16 + r + ((lane >> 4) << 3);
      int gcol = tj * 16 + (lane & 15);
      gp[grow * C_CH + gcol] = acc[tj][r];
    }
  if (tid < C_CH) spart[p * C_CH + tid] = ssum;
}

// ---------------- Kernel 2: deterministic fixed-order reduction ----------------
__global__ void __launch_bounds__(256) k_reduce(const float* __restrict__ gpart,
                                                const float* __restrict__ spart,
                                                float* __restrict__ gram,
                                                float* __restrict__ sums) {
  int t = blockIdx.x * blockDim.x + threadIdx.x;  // 0..16383
  float a = 0.0f;
  for (int p = 0; p < 128; ++p) a += gpart[(size_t)p * (C_CH * C_CH) + t];
  gram[t] = a;
  if (t < C_CH) {
    float s = 0.0f;
    for (int p = 0; p < 128; ++p) s += spart[p * C_CH + t];
    sums[t] = s;
  }
}

// 128x128 matmul in LDS: D = A*B, wave wv computes row-block wv (8 tiles)
__device__ __forceinline__ void mm128(float* D, const float* A, const float* Bm,
                                      int wv, int lane) {
  v8f acc[8];
#pragma unroll
  for (int t = 0; t < 8; ++t) acc[t] = v8f_zero();
  for (int k = 0; k < 128; k += 4) {
    v2f a = loadA(A, MS, wv * 16, k, lane);
#pragma unroll
    for (int tj = 0; tj < 8; ++tj) {
      v2f b   = loadB_rm(Bm, MS, k, tj * 16, lane);
      acc[tj] = wmma_f32_k4(a, b, acc[tj]);
    }
  }
#pragma unroll
  for (int tj = 0; tj < 8; ++tj) storeC(D, MS, wv * 16, tj * 16, lane, acc[tj]);
}

// ---------------- Kernel 3: Sigma + Newton-Schulz + M = rot*wm, b = M*mean ----------------
// single workgroup, 256 threads, dynamic LDS: 4 * 128x132 matrices + mean + scratch
__global__ void __launch_bounds__(256) k_newton(const float* __restrict__ rot,
                                                const float* __restrict__ gram,
                                                const float* __restrict__ sums,
                                                float* __restrict__ Mout,
                                                float* __restrict__ bout) {
  extern __shared__ float sm[];
  float* S     = sm;               // SigmaN
  float* P     = S + C_CH * MS;    // Newton iterate
  float* T1    = P + C_CH * MS;
  float* T2    = T1 + C_CH * MS;
  float* meanv = T2 + C_CH * MS;   // 128
  float* red   = meanv + C_CH;     // scratch

  const int tid  = threadIdx.x;
  const int lane = tid & 31;
  const int wv   = tid >> 5;
  const float invm = 1.0f / (float)M_TOT;

  if (tid < C_CH) meanv[tid] = sums[tid] * invm;
  __syncthreads();

  // Sigma = eps*I + Gram/m - mean*mean^T  (unnormalized, into S)
  for (int i = tid; i < C_CH * C_CH; i += 256) {
    int r = i >> 7, c = i & 127;
    S[r * MS + c] =
        gram[i] * invm - meanv[r] * meanv[c] + ((r == c) ? EPS_ : 0.0f);
  }
  __syncthreads();
  if (tid == 0) {
    float tr = 0.0f;
    for (int c = 0; c < C_CH; ++c) tr += S[c * MS + c];
    red[0] = 1.0f / tr;  // rTr
  }
  __syncthreads();
  const float rTr = red[0];
  for (int i = tid; i < C_CH * C_CH; i += 256) {
    int r = i >> 7, c = i & 127;
    S[r * MS + c] *= rTr;                      // Sigma_N
    P[r * MS + c] = (r == c) ? 1.0f : 0.0f;    // P = I
  }
  __syncthreads();

  for (int it = 0; it < NEWTON_T; ++it) {
    mm128(T1, P, P, wv, lane);   // T1 = P^2
    __syncthreads();
    mm128(T2, T1, P, wv, lane);  // T2 = P^3
    __syncthreads();
    // D = T2 * SigmaN ; P = 1.5P - 0.5D  (each wave updates its own row-block)
    v8f acc[8];
#pragma unroll
    for (int t = 0; t < 8; ++t) acc[t] = v8f_zero();
    for (int k = 0; k < 128; k += 4) {
      v2f a = loadA(T2, MS, wv * 16, k, lane);
#pragma unroll
      for (int tj = 0; tj < 8; ++tj) {
        v2f b   = loadB_rm(S, MS, k, tj * 16, lane);
        acc[tj] = wmma_f32_k4(a, b, acc[tj]);
      }
    }
#pragma unroll
    for (int tj = 0; tj < 8; ++tj)
#pragma unroll
      for (int r = 0; r < 8; ++r) {
        int row = wv * 16 + r + ((lane >> 4) << 3);
        int col = tj * 16 + (lane & 15);
        float pv = P[row * MS + col];
        P[row * MS + col] = 1.5f * pv - 0.5f * acc[tj][r];
      }
    __syncthreads();
  }

  // M = sqrt(rTr) * (rot @ P) ; b = M @ mean
  for (int i = tid; i < C_CH * C_CH; i += 256)
    T1[(i >> 7) * MS + (i & 127)] = rot[i];
  __syncthreads();
  mm128(T2, T1, P, wv, lane);
  __syncthreads();
  const float sr = sqrtf(rTr);
  for (int i = tid; i < C_CH * C_CH; i += 256)
    Mout[i] = T2[(i >> 7) * MS + (i & 127)] * sr;
  if (tid < C_CH) {
    float bb = 0.0f;
    for (int c = 0; c < C_CH; ++c) bb += T2[tid * MS + c] * sr * meanv[c];
    bout[tid] = bb;
  }
}

// ---------------- Kernel 4: out = M @ x - b ----------------
// grid (64 batches, 7 segments); each WG caches M in LDS, streams x tiles async.
__global__ void __launch_bounds__(256) k_apply(const float* __restrict__ X,
                                               const float* __restrict__ Mm,
                                               const float* __restrict__ bias,
                                               float* __restrict__ out) {
  extern __shared__ float sm[];
  float* ldsM = sm;                  // 128 x 132
  float* xt   = ldsM + C_CH * MS;    // 128 x 68
  float* bsh  = xt + C_CH * XS;      // 128

  const int b    = blockIdx.x;
  const int seg  = blockIdx.y;
  const int tid  = threadIdx.x;
  const int lane = tid & 31;
  const int wv   = tid >> 5;

  // load M into LDS (async, 16B granules)
  for (int i = tid; i < (C_CH * C_CH) / 4; i += 256) {
    int row  = i >> 5;         // 32 float4 per row
    int col4 = (i & 31) * 4;
    async_cp16(&ldsM[row * MS + col4], Mm + row * C_CH + col4);
  }
  if (tid < C_CH) bsh[tid] = bias[tid];
  wait_async_copies();
  __syncthreads();

  const float* Xb  = X + (size_t)b * C_CH * HW_;
  float*       Ob  = out + (size_t)b * C_CH * HW_;
  const int    row = tid >> 1;
  const int    mo  = (tid & 1) * 32;

  for (int ch = 0; ch < 7; ++ch) {
    const int    hw0 = (seg * 7 + ch) * CHUNK;
    const float* g   = Xb + (size_t)row * HW_ + hw0 + mo;
    float*       l   = &xt[row * XS + mo];
#pragma unroll
    for (int q = 0; q < 8; ++q) async_cp16(l + q * 4, g + q * 4);
    if (ch + 1 < 7)
      __builtin_prefetch(Xb + (size_t)row * HW_ + hw0 + CHUNK + mo, 0, 0);
    wait_async_copies();
    __syncthreads();

    v8f acc[4];
#pragma unroll
    for (int t = 0; t < 4; ++t) acc[t] = v8f_zero();
    for (int k = 0; k < C_CH; k += 4) {
      v2f a = loadA(ldsM, MS, wv * 16, k, lane);
#pragma unroll
      for (int tm = 0; tm < 4; ++tm) {
        v2f bb   = loadB_rm(xt, XS, k, tm * 16, lane);
        acc[tm]  = wmma_f32_k4(a, bb, acc[tm]);
      }
    }
#pragma unroll
    for (int tm = 0; tm < 4; ++tm)
#pragma unroll
      for (int r = 0; r < 8; ++r) {
        int d = wv * 16 + r + ((lane >> 4) << 3);
        int m = hw0 + tm * 16 + (lane & 15);
        Ob[(size_t)d * HW_ + m] = acc[tm][r] - bsh[d];
      }
    __syncthreads();
  }
}

extern "C" void kernel_launch(void* const* d_in, const int* in_sizes, int n_in,
                              void* d_out, int out_size, void* d_ws,
                              size_t ws_size, hipStream_t stream) {
  (void)in_sizes; (void)n_in; (void)out_size; (void)ws_size;
  const float* X   = (const float*)d_in[0];
  const float* rot = (const float*)d_in[1];
  float*       out = (float*)d_out;
  float*       ws  = (float*)d_ws;

  // workspace layout (floats)
  float* gpart = ws;                               // 128 * 16384
  float* spart = gpart + (size_t)128 * 16384;      // 128 * 128
  float* gram  = spart + (size_t)128 * 128;        // 16384
  float* sums  = gram + 16384;                     // 128
  float* Mm    = sums + 128;                       // 16384
  float* bias  = Mm + 16384;                       // 128

  dim3 blk(256);
  k_gram<<<dim3(128), blk, 0, stream>>>(X, gpart, spart);
  k_reduce<<<dim3(64), blk, 0, stream>>>(gpart, spart, gram, sums);
  size_t shm2 = (size_t)(4 * C_CH * MS + 256) * sizeof(float);
  k_newton<<<dim3(1), blk, shm2, stream>>>(rot, gram, sums, Mm, bias);
  size_t shm3 = (size_t)(C_CH * MS + C_CH * XS + 128) * sizeof(float);
  k_apply<<<dim3(64, 7), blk, shm3, stream>>>(X, Mm, bias, out);
}